// test_class_83313775608314
// MI455X (gfx1250) — compile-verified
//
#include <hip/hip_runtime.h>
#include <hip/hip_bf16.h>
#include <math.h>

// ---------------- problem constants (match reference) ----------------
#define A_      8732
#define C_      91
#define B_      32
#define NCLS    90              // foreground classes
#define TOPK_   400
#define NCAND   (NCLS * TOPK_)  // 36000 candidates per image
#define MAXDET  200
#define IMG_    300.0f
#define SCORE_THRESH 0.01f
#define NMS_THRESH   0.45f
#define BBOX_CLIP    4.135166556742356f  // log(1000/16)
#define NBINS   1024

#define NEG_INF (-__builtin_inff())

typedef unsigned int u32x4 __attribute__((ext_vector_type(4)));
typedef int          i32x8 __attribute__((ext_vector_type(8)));
typedef int          i32x4 __attribute__((ext_vector_type(4)));

// =====================================================================
// Kernel A: per-(b,anchor) box decode + softmax; optionally writes the
// transposed masked score tensor [B, NCLS, A] (coalesced stores) so the
// top-k kernel can stream it instead of gathering strided logits.
// logits (102MB) are read 2-3x; repeat passes hit the 192MB global L2.
// =====================================================================
__global__ void decode_norm_kernel(const float* __restrict__ logits,
                                   const float* __restrict__ rel,
                                   const float* __restrict__ anchors,
                                   float* __restrict__ ws_norm,   // [B*A, 2]
                                   float* __restrict__ ws_boxes,  // [B*A, 4]
                                   float* __restrict__ scores_t)  // [B, NCLS, A] or null
{
    int idx = blockIdx.x * blockDim.x + threadIdx.x;
    if (idx >= B_ * A_) return;
    int b = idx / A_;
    int a = idx - b * A_;

    const float* lg = logits + (size_t)idx * C_;
    __builtin_prefetch(lg, 0, 0);  // -> global_prefetch_b8 (gfx1250)

    float mx = NEG_INF;
    #pragma unroll 7
    for (int c = 0; c < C_; ++c) mx = fmaxf(mx, lg[c]);
    float se = 0.f;
    #pragma unroll 7
    for (int c = 0; c < C_; ++c) se += __expf(lg[c] - mx);
    ws_norm[(size_t)idx * 2 + 0] = mx;
    ws_norm[(size_t)idx * 2 + 1] = se;

    if (scores_t) {
        float inv = 1.0f / se;
        float* dst = scores_t + (size_t)b * NCLS * A_ + a;
        #pragma unroll 6
        for (int c = 1; c < C_; ++c) {
            float sc = __expf(lg[c] - mx) * inv;
            dst[(size_t)(c - 1) * A_] = (sc > SCORE_THRESH) ? sc : NEG_INF;
        }
    }

    // box decode (weights 10,10,5,5), clip to [0, 300]
    const float* an = anchors + (size_t)a * 4;
    float ax0 = an[0], ay0 = an[1], ax1 = an[2], ay1 = an[3];
    float wa = ax1 - ax0, ha = ay1 - ay0;
    float cxa = ax0 + 0.5f * wa, cya = ay0 + 0.5f * ha;
    const float* r = rel + (size_t)idx * 4;
    float dx = r[0] * 0.1f;
    float dy = r[1] * 0.1f;
    float dw = fminf(r[2] * 0.2f, BBOX_CLIP);
    float dh = fminf(r[3] * 0.2f, BBOX_CLIP);
    float cx = dx * wa + cxa, cy = dy * ha + cya;
    float w  = __expf(dw) * wa, h = __expf(dh) * ha;
    float x0 = fminf(fmaxf(cx - 0.5f * w, 0.f), IMG_);
    float y0 = fminf(fmaxf(cy - 0.5f * h, 0.f), IMG_);
    float x1 = fminf(fmaxf(cx + 0.5f * w, 0.f), IMG_);
    float y1 = fminf(fmaxf(cy + 0.5f * h, 0.f), IMG_);
    float4 bx; bx.x = x0; bx.y = y0; bx.z = x1; bx.w = y1;
    ((float4*)ws_boxes)[idx] = bx;
}

// ---------------------------------------------------------------------
// shared top-400 selection body: s_sc[A_] holds masked scores in LDS.
// histogram cutoff over softmax's (0,1) range, then atomic compaction.
// ---------------------------------------------------------------------
__device__ __forceinline__ void topk_select_body(
    float* s_sc, unsigned* s_hist, unsigned* s_cut, unsigned* s_cnt,
    int b, int c, int cls,
    const float* __restrict__ ws_boxes,
    float* __restrict__ cand_sc, float* __restrict__ cand_box)
{
    size_t base = (size_t)b * A_;
    for (int a = threadIdx.x; a < A_; a += blockDim.x) {
        float m = s_sc[a];
        if (m != NEG_INF) {
            int bin = (int)(m * (float)NBINS);
            bin = bin < 0 ? 0 : (bin > NBINS - 1 ? NBINS - 1 : bin);
            atomicAdd(&s_hist[bin], 1u);
        }
    }
    // init all TOPK candidate score slots for this (b,c) to -inf
    float* csc = cand_sc + (size_t)b * NCAND + (size_t)c * TOPK_;
    for (int p = threadIdx.x; p < TOPK_; p += blockDim.x) csc[p] = NEG_INF;
    __syncthreads();

    if (threadIdx.x == 0) {
        unsigned cum = 0; int cut = 0;
        for (int i = NBINS - 1; i >= 0; --i) {
            cum += s_hist[i];
            if (cum >= TOPK_) { cut = i; break; }
        }
        *s_cut = (unsigned)cut;
    }
    __syncthreads();

    int   cut = (int)*s_cut;
    float off = (float)cls * (IMG_ + 1.0f);
    float4* cbox = (float4*)cand_box + (size_t)b * NCAND + (size_t)c * TOPK_;

    for (int a = threadIdx.x; a < A_; a += blockDim.x) {
        float m = s_sc[a];
        if (m != NEG_INF) {
            int bin = (int)(m * (float)NBINS);
            bin = bin < 0 ? 0 : (bin > NBINS - 1 ? NBINS - 1 : bin);
            if (bin >= cut) {
                unsigned pos = atomicAdd(s_cnt, 1u);
                if (pos < TOPK_) {
                    csc[pos] = m;
                    float4 bx = ((const float4*)ws_boxes)[base + a];
                    bx.x += off; bx.y += off; bx.z += off; bx.w += off;
                    cbox[pos] = bx;
                }
            }
        }
    }
}

// =====================================================================
// Kernel B (fast path): per-(b,class) top-400. The contiguous 34,928B
// score row is DMA'd into LDS by the Tensor Data Mover: one wave issues
// a single tensor_load_to_lds (1D tile D#), waits on TENSORcnt, then the
// block barriers. Replaces ~8.7K per-lane loads with one DMA descriptor.
// =====================================================================
__global__ void topk_scores_kernel(const float* __restrict__ scores_t,
                                   const float* __restrict__ ws_boxes,
                                   float* __restrict__ cand_sc,    // [B, NCAND]
                                   float* __restrict__ cand_box)   // [B, NCAND, 4]
{
    __shared__ float    s_sc[A_];
    __shared__ unsigned s_hist[NBINS];
    __shared__ unsigned s_cut, s_cnt;

    int b   = blockIdx.x / NCLS;
    int c   = blockIdx.x % NCLS;
    int cls = c + 1;

    const float* row = scores_t + ((size_t)b * NCLS + c) * A_;

    // ---- TDM: DMA the whole score row into LDS (wave 0 only; TDM ignores EXEC,
    //      the branch ensures a single wave issues a single descriptor) ----
    if (threadIdx.x < 32) {
        unsigned long long ga = (unsigned long long)(size_t)row;
        unsigned lds_base = (unsigned)(size_t)&s_sc[0];
        // D# group0: count=1 | lds_addr | global_addr(57b) | type=2
        u32x4 g0;
        g0.x = 1u;                                             // count=1, user mode
        g0.y = lds_base;                                       // bits 63:32
        g0.z = (unsigned)(ga & 0xFFFFFFFFull);                 // bits 95:64
        g0.w = (unsigned)((ga >> 32) & 0x1FFFFFFull) | (2u << 30); // addr[56:32] | type=2
        // D# group1: data_size=4B, tensor_dim0=tile_dim0=A, stride0=A
        i32x8 g1;
        g1.s0 = (int)(2u << 16);            // data_size=2 (4 bytes); mask/flags 0
        g1.s1 = (int)((A_ & 0xFFFF) << 16); // tensor_dim0[15:0] @ bits 63:48
        g1.s2 = (int)(A_ >> 16);            // tensor_dim0[31:16]; tensor_dim1=0
        g1.s3 = (int)(A_ << 16);            // tile_dim0 @ bits 127:112
        g1.s4 = 0;                          // tile_dim1=0, tile_dim2=0 (1D tile)
        g1.s5 = (int)A_;                    // tensor_dim0_stride[31:0]
        g1.s6 = 0;
        g1.s7 = 0;
        i32x4 gz = {0, 0, 0, 0};            // groups 2/3 unused (<=2D)
        asm volatile("tensor_load_to_lds %0, %1, %2, %3"
                     :: "s"(g0), "s"(g1), "s"(gz), "s"(gz) : "memory");
        __builtin_amdgcn_s_wait_tensorcnt(0);
    }
    for (int i = threadIdx.x; i < NBINS; i += blockDim.x) s_hist[i] = 0u;
    if (threadIdx.x == 0) s_cnt = 0u;
    __syncthreads();   // TDM done (wave0 waited) + hist zeroed -> LDS visible

    topk_select_body(s_sc, s_hist, &s_cut, &s_cnt, b, c, cls,
                     ws_boxes, cand_sc, cand_box);
}

// =====================================================================
// Kernel B (fallback, small workspace): recompute per-class scores from
// logits + per-anchor normalizer (strided L2 gather).
// =====================================================================
__global__ void topk_logits_kernel(const float* __restrict__ logits,
                                   const float* __restrict__ ws_norm,
                                   const float* __restrict__ ws_boxes,
                                   float* __restrict__ cand_sc,
                                   float* __restrict__ cand_box)
{
    __shared__ float    s_sc[A_];
    __shared__ unsigned s_hist[NBINS];
    __shared__ unsigned s_cut, s_cnt;

    int b   = blockIdx.x / NCLS;
    int c   = blockIdx.x % NCLS;
    int cls = c + 1;

    for (int i = threadIdx.x; i < NBINS; i += blockDim.x) s_hist[i] = 0u;
    if (threadIdx.x == 0) s_cnt = 0u;

    size_t base = (size_t)b * A_;
    for (int a = threadIdx.x; a < A_; a += blockDim.x) {
        float l  = logits[(base + a) * C_ + cls];
        float mx = ws_norm[(base + a) * 2 + 0];
        float se = ws_norm[(base + a) * 2 + 1];
        float sc = __expf(l - mx) / se;
        s_sc[a] = (sc > SCORE_THRESH) ? sc : NEG_INF;
    }
    __syncthreads();

    topk_select_body(s_sc, s_hist, &s_cut, &s_cnt, b, c, cls,
                     ws_boxes, cand_sc, cand_box);
}

// =====================================================================
// Kernel C: greedy NMS, one 1024-thread (32-wave) block per image.
// 36000 candidate scores live in 144KB dynamic LDS (CDNA5: 320KB/WGP),
// staged via the gfx1250 async-to-LDS path (ASYNCcnt-tracked).
// Candidate boxes (18MB total) are read from L2 each round.
// =====================================================================
__global__ __launch_bounds__(1024)
void nms_kernel(const float* __restrict__ cand_sc,
                const float* __restrict__ cand_box,
                float* __restrict__ out_boxes,   // [B, MAXDET, 4]
                float* __restrict__ out_scores,  // [B, MAXDET]
                float* __restrict__ out_labels)  // [B, MAXDET] (as float)
{
    extern __shared__ float smem[];
    float* s_sc   = smem;                      // NCAND scores
    float* s_rv   = smem + NCAND;              // 32 wave-partial maxima
    int*   s_ri   = (int*)(smem + NCAND + 32); // 32 wave-partial argmax
    float* s_best = smem + NCAND + 64;         // [0..3]=box, [4]=score, [5]=idx(bits)

    int b = blockIdx.x;
    const float*  gsc  = cand_sc + (size_t)b * NCAND;
    const float4* gbox = (const float4*)cand_box + (size_t)b * NCAND;

    // ---- async stage: global scores -> LDS (b128 = 4 floats per issue) ----
    for (int i = threadIdx.x; i < NCAND / 4; i += blockDim.x) {
        unsigned lds_off = (unsigned)(size_t)(&s_sc[4 * i]);
        const float* g = gsc + 4 * i;
        asm volatile("global_load_async_to_lds_b128 %0, %1, off"
                     :: "v"(lds_off), "v"(g) : "memory");
    }
    asm volatile("s_wait_asynccnt 0" ::: "memory");
    __syncthreads();

    int lane = threadIdx.x & 31;
    int wid  = threadIdx.x >> 5;

    for (int it = 0; it < MAXDET; ++it) {
        // ---- parallel argmax over NCAND ----
        float v  = NEG_INF;
        int   vi = -1;
        for (int j = threadIdx.x; j < NCAND; j += blockDim.x) {
            float s = s_sc[j];
            if (s > v) { v = s; vi = j; }
        }
        #pragma unroll
        for (int o = 16; o > 0; o >>= 1) {  // wave32 shuffle reduction
            float ov = __shfl_down(v, o, 32);
            int   oi = __shfl_down(vi, o, 32);
            if (ov > v) { v = ov; vi = oi; }
        }
        if (lane == 0) { s_rv[wid] = v; s_ri[wid] = vi; }
        __syncthreads();
        if (wid == 0) {
            float vv = s_rv[lane];
            int   ii = s_ri[lane];
            #pragma unroll
            for (int o = 16; o > 0; o >>= 1) {
                float ov = __shfl_down(vv, o, 32);
                int   oi = __shfl_down(ii, o, 32);
                if (ov > vv) { vv = ov; ii = oi; }
            }
            if (lane == 0) {
                s_best[4] = vv;
                ((int*)s_best)[5] = ii;
                if (ii >= 0 && vv != NEG_INF) {
                    float4 bb = gbox[ii];
                    s_best[0] = bb.x; s_best[1] = bb.y;
                    s_best[2] = bb.z; s_best[3] = bb.w;
                }
            }
        }
        __syncthreads();

        float bs = s_best[4];
        int   bi = ((int*)s_best)[5];
        bool valid = (bi >= 0) && (bs != NEG_INF);   // uniform across block

        if (valid) {
            float bx0 = s_best[0], by0 = s_best[1];
            float bx1 = s_best[2], by1 = s_best[3];
            float barea = (bx1 - bx0) * (by1 - by0);
            for (int j = threadIdx.x; j < NCAND; j += blockDim.x) {
                if (s_sc[j] != NEG_INF) {
                    float4 ob = gbox[j];
                    float lx = fmaxf(bx0, ob.x), ly = fmaxf(by0, ob.y);
                    float rx = fminf(bx1, ob.z), ry = fminf(by1, ob.w);
                    float iw = fmaxf(rx - lx, 0.f), ih = fmaxf(ry - ly, 0.f);
                    float inter = iw * ih;
                    float oarea = (ob.z - ob.x) * (ob.w - ob.y);
                    float iou = inter / (barea + oarea - inter + 1e-9f);
                    if (iou > NMS_THRESH) s_sc[j] = NEG_INF;
                }
            }
            if (threadIdx.x == 0) s_sc[bi] = NEG_INF;  // self-suppress
        }

        if (threadIdx.x == 0) {
            float* ob = out_boxes + ((size_t)b * MAXDET + it) * 4;
            if (valid) {
                int   label = bi / TOPK_ + 1;
                float off   = (float)label * (IMG_ + 1.0f);
                ob[0] = s_best[0] - off; ob[1] = s_best[1] - off;
                ob[2] = s_best[2] - off; ob[3] = s_best[3] - off;
                out_scores[(size_t)b * MAXDET + it] = bs;
                out_labels[(size_t)b * MAXDET + it] = (float)label;
            } else {
                ob[0] = 0.f; ob[1] = 0.f; ob[2] = 0.f; ob[3] = 0.f;
                out_scores[(size_t)b * MAXDET + it] = 0.f;
                out_labels[(size_t)b * MAXDET + it] = 0.f;
            }
        }
        __syncthreads();
    }
}

// =====================================================================
// launcher
// =====================================================================
extern "C" void kernel_launch(void* const* d_in, const int* in_sizes, int n_in,
                              void* d_out, int out_size, void* d_ws, size_t ws_size,
                              hipStream_t stream)
{
    const float* logits  = (const float*)d_in[0];  // [B, A, C]
    const float* rel     = (const float*)d_in[1];  // [B, A, 4]
    const float* anchors = (const float*)d_in[2];  // [A, 4]

    // workspace layout (floats), all 16B aligned
    float* ws       = (float*)d_ws;
    float* ws_norm  = ws;                                       // B*A*2
    float* ws_boxes = ws_norm + (size_t)B_ * A_ * 2;            // B*A*4
    float* cand_sc  = ws_boxes + (size_t)B_ * A_ * 4;           // B*NCAND
    float* cand_box = cand_sc + (size_t)B_ * NCAND;             // B*NCAND*4
    float* scores_t = cand_box + (size_t)B_ * NCAND * 4;        // B*NCLS*A (optional)

    size_t base_elems   = (size_t)B_ * A_ * 6 + (size_t)B_ * NCAND * 5;
    size_t scoret_elems = (size_t)B_ * NCLS * A_;
    bool use_transposed = ws_size >= (base_elems + scoret_elems) * sizeof(float);

    float* out        = (float*)d_out;
    float* out_boxes  = out;                                    // B*MAXDET*4
    float* out_scores = out + (size_t)B_ * MAXDET * 4;          // B*MAXDET
    float* out_labels = out_scores + (size_t)B_ * MAXDET;       // B*MAXDET

    int total = B_ * A_;
    decode_norm_kernel<<<(total + 255) / 256, 256, 0, stream>>>(
        logits, rel, anchors, ws_norm, ws_boxes,
        use_transposed ? scores_t : (float*)nullptr);

    if (use_transposed) {
        topk_scores_kernel<<<B_ * NCLS, 256, 0, stream>>>(
            scores_t, ws_boxes, cand_sc, cand_box);
    } else {
        topk_logits_kernel<<<B_ * NCLS, 256, 0, stream>>>(
            logits, ws_norm, ws_boxes, cand_sc, cand_box);
    }

    size_t smem_bytes = (size_t)(NCAND + 72) * sizeof(float);   // ~144.3 KB (< 320 KB WGP LDS)
    hipFuncSetAttribute((const void*)nms_kernel,
                        hipFuncAttributeMaxDynamicSharedMemorySize,
                        (int)smem_bytes);
    nms_kernel<<<B_, 1024, smem_bytes, stream>>>(
        cand_sc, cand_box, out_boxes, out_scores, out_labels);
}